// Communication_76802605187251
// MI455X (gfx1250) — compile-verified
//
#include <hip/hip_runtime.h>
#include <hip/hip_bf16.h>

typedef __attribute__((ext_vector_type(2))) float v2f;
typedef __attribute__((ext_vector_type(8))) float v8f;

namespace {

constexpr int kB = 16, kL = 6, kH = 352, kW = 704;
constexpr int kMaskN = kB * kL * kH * kW;   // 23,789,568
constexpr float kThresh = 0.01f;

// Separable 1-D Gaussian factor of the reference's (1/(2*pi*sigma)) * exp(-(x^2+y^2)/(2*sigma^2))
// with sigma = 1:  g1d[k] = exp(-(k-2)^2/2) / sqrt(2*pi); outer product == reference 2D kernel.
// Banded matrix entry G[r][c] = g1d[r-c] for 0 <= r-c <= 4 else 0.
__device__ __forceinline__ float gband(int r, int c) {
    int d = r - c;
    float v = 0.0f;
    v = (d == 0 || d == 4) ? 0.05399096651318806f : v;
    v = (d == 1 || d == 3) ? 0.24197072451914337f : v;
    v = (d == 2)           ? 0.39894228040143270f : v;
    return v;
}

// Block: 256 threads = 8 waves; wave w -> 16x16 output tile at (th = w>>2, tw = w&3)
// Block output region: 32 (H) x 64 (W). Grid: (704/64, 352/32, B*L) = (11, 11, 96).
__global__ __launch_bounds__(256) void comm_kernel(const float* __restrict__ x,
                                                   float* __restrict__ out,
                                                   unsigned* __restrict__ cnt_accum) {
    __shared__ float clds[36][68];      // conf with 2-halo, stride 68 (bank-friendly)
    __shared__ float tmplds[8][20 * 17];// per-wave horizontal-pass scratch, stride 17
    __shared__ unsigned scnt;

    const int tid = threadIdx.x;
    const int bl = blockIdx.z;               // fused (b*L + l)
    const int h0 = blockIdx.y * 32;
    const int w0 = blockIdx.x * 64;
    if (tid == 0) scnt = 0;

    // ---- Stage 1: conf = sigmoid(max over C) into LDS (zero outside image = conv zero-pad)
    const float* xp = x + (size_t)bl * 2u * kH * kW;
    for (int i = tid; i < 36 * 68; i += 256) {
        int rr = i / 68, cc = i % 68;
        int h = h0 - 2 + rr, w = w0 - 2 + cc;
        float v = 0.0f;
        if (h >= 0 && h < kH && w >= 0 && w < kW) {
            size_t o = (size_t)h * kW + w;
            float m = fmaxf(xp[o], xp[(size_t)kH * kW + o]);
            v = 1.0f / (1.0f + __expf(-m));   // sigmoid(max) == max(sigmoid)
        }
        clds[rr][cc] = v;
    }
    __syncthreads();

    const int wave = tid >> 5, lane = tid & 31;
    const int half = lane >> 4, n = lane & 15;
    const int th = wave >> 2, tw = wave & 3;
    const int rbase = 16 * th, cbase = 16 * tw;

    // ---- Stage 2: horizontal pass  tmp(20x16) = P(20x20) * Gh(20x16), K in chunks of 4.
    // A layout (16x4 f32): lanes 0-15 K={0,1}, lanes 16-31 K={2,3} (VGPR0/1).
    // B layout (4x16 f32): lane n = column; VGPR0/1 = K rows {0,1} (half 0) / {2,3} (half 1).
    v8f acc0 = {0, 0, 0, 0, 0, 0, 0, 0};  // patch rows 0..15
    v8f acc1 = {0, 0, 0, 0, 0, 0, 0, 0};  // patch rows 16..19 (M 0..3 valid)
#pragma unroll
    for (int j = 0; j < 5; ++j) {
        const int cc = 4 * j + 2 * half;
        v2f a0, a1, bg;
        a0.x = clds[rbase + n][cbase + cc];
        a0.y = clds[rbase + n][cbase + cc + 1];
        if (n < 4) {
            a1.x = clds[rbase + 16 + n][cbase + cc];
            a1.y = clds[rbase + 16 + n][cbase + cc + 1];
        } else {
            a1.x = 0.0f; a1.y = 0.0f;      // rows >= 20 of chunk-1 are never consumed
        }
        bg.x = gband(cc, n);
        bg.y = gband(cc + 1, n);
        acc0 = __builtin_amdgcn_wmma_f32_16x16x4_f32(false, a0, false, bg, (short)0, acc0, false, false);
        acc1 = __builtin_amdgcn_wmma_f32_16x16x4_f32(false, a1, false, bg, (short)0, acc1, false, false);
    }

    // D layout (16x16 f32): VGPR v -> row v (lanes 0-15) / row v+8 (lanes 16-31), col = n.
    float* tws = tmplds[wave];
#pragma unroll
    for (int v = 0; v < 8; ++v) tws[(v + 8 * half) * 17 + n] = acc0[v];
    if (half == 0) {
#pragma unroll
        for (int v = 0; v < 4; ++v) tws[(16 + v) * 17 + n] = acc1[v];
    }
    __syncthreads();

    // ---- Stage 3: vertical pass  out(16x16) = Gv(16x20) * tmp(20x16)
    v8f dv = {0, 0, 0, 0, 0, 0, 0, 0};
#pragma unroll
    for (int j = 0; j < 5; ++j) {
        const int r = 4 * j + 2 * half;
        v2f a, bb;
        a.x = gband(r, n);                 // Gv[m=n][r]
        a.y = gband(r + 1, n);
        bb.x = tws[r * 17 + n];
        bb.y = tws[(r + 1) * 17 + n];
        dv = __builtin_amdgcn_wmma_f32_16x16x4_f32(false, a, false, bb, (short)0, dv, false, false);
    }

    // ---- Stage 4: mask + raw writes, hierarchical count for rate
    const int l = bl % kL;
    const int gw = w0 + cbase + n;
    unsigned cnt = 0;
    float* masks = out;
    float* raw   = out + kMaskN + 1;
#pragma unroll
    for (int v = 0; v < 8; ++v) {
        const int gh = h0 + rbase + v + 8 * half;
        const float s = dv[v];
        const bool over = s > kThresh;     // count BEFORE ego forcing (matches reference order)
        cnt += over ? 1u : 0u;
        const size_t idx = ((size_t)bl * kH + gh) * (size_t)kW + gw;
        masks[idx] = (l == 0) ? 1.0f : (over ? 1.0f : 0.0f);
        raw[idx]   = s;
    }
    atomicAdd(&scnt, cnt);
    __syncthreads();
    if (tid == 0) atomicAdd(cnt_accum, scnt);
}

__global__ void zero_cnt(unsigned* c) {
    if (threadIdx.x == 0) *c = 0u;
}

__global__ void write_rate(const unsigned* __restrict__ c, float* __restrict__ out) {
    if (threadIdx.x == 0) {
        // mean over B of (per-batch count / (L*H*W)) == total / (B*L*H*W)
        out[kMaskN] = (float)((double)(*c) / (double)kMaskN);
    }
}

} // namespace

extern "C" void kernel_launch(void* const* d_in, const int* in_sizes, int n_in,
                              void* d_out, int out_size, void* d_ws, size_t ws_size,
                              hipStream_t stream) {
    (void)in_sizes; (void)n_in; (void)out_size; (void)ws_size;
    const float* x = (const float*)d_in[0];  // d_in[1] (gauss_w) reproduced analytically
    float* out = (float*)d_out;
    unsigned* cnt = (unsigned*)d_ws;

    zero_cnt<<<1, 1, 0, stream>>>(cnt);
    dim3 grid(kW / 64, kH / 32, kB * kL);    // (11, 11, 96)
    comm_kernel<<<grid, 256, 0, stream>>>(x, out, cnt);
    write_rate<<<1, 1, 0, stream>>>(cnt, out);
}